// VQ_16484084482616
// MI455X (gfx1250) — compile-verified
//
#include <hip/hip_runtime.h>

typedef __attribute__((ext_vector_type(2))) float v2f;
typedef __attribute__((ext_vector_type(8))) float v8f;

#define N_IMG      32
#define C_DIM      256
#define HW         1024                 // H*W
#define D_DIM      1024
#define M_TOTAL    (N_IMG * HW)         // 32768 pixels
#define PIX_PER_WG 128
#define THREADS    256
#define D_TILES    (D_DIM / 16)         // 64
#define K_STEPS    (C_DIM / 4)          // 64 wmma per 16x16 tile
#define RSTR       258                  // padded LDS row stride (floats)

// LDS layout (float offsets)
#define X_OFF      0
#define X_SIZE     (PIX_PER_WG * RSTR)  // 33024
#define DICT_OFF   (X_OFF + X_SIZE)
#define DICT_SIZE  (16 * RSTR)          // 4128
#define DN_OFF     (DICT_OFF + DICT_SIZE)
#define IDX_OFF    (DN_OFF + 16)
#define SMEM_FLOATS (IDX_OFF + PIX_PER_WG)
#define SMEM_BYTES  (SMEM_FLOATS * 4)   // ~149 KB

#define NCHW ((size_t)N_IMG * C_DIM * HW)

extern "C" __global__ __launch_bounds__(THREADS)
void vq_wmma_kernel(const float* __restrict__ x,
                    const float* __restrict__ dict,
                    float* __restrict__ out)
{
    extern __shared__ float smem[];
    float* xLds    = smem + X_OFF;
    float* dictLds = smem + DICT_OFF;
    float* dnLds   = smem + DN_OFF;
    int*   idxLds  = (int*)(smem + IDX_OFF);

    const int tid  = threadIdx.x;
    const int lane = tid & 31;
    const int wave = tid >> 5;
    const int wgPixBase = blockIdx.x * PIX_PER_WG;
    const int img = wgPixBase >> 10;          // 128 | 1024, so one image per WG
    const int hw0 = wgPixBase & (HW - 1);

    // ---- Stage x tile [128 pixels][256 ch] into LDS (transpose NCHW -> pix-major)
    for (int j = tid; j < PIX_PER_WG * C_DIM; j += THREADS) {
        int c = j >> 7;                   // j / 128
        int p = j & (PIX_PER_WG - 1);
        xLds[p * RSTR + c] = x[(size_t)img * C_DIM * HW + (size_t)c * HW + hw0 + p];
    }

    const int m     = lane & 15;          // A: M row, B: N col (identical index)
    const int khalf = lane >> 4;          // K-pair selector within a 4-K group
    const float* aRow = xLds + (wave * 16 + m) * RSTR;
    const float* bRow = dictLds + m * RSTR;

    float best[8];
    int   bidx[8];
#pragma unroll
    for (int r = 0; r < 8; ++r) { best[r] = 3.4e38f; bidx[r] = 0; }

    for (int dt = 0; dt < D_TILES; ++dt) {
        __syncthreads();                  // previous tile's compute done
        const int dBase = dt * 16;
        if (dt + 1 < D_TILES)             // prefetch next dict tile (64B/thread)
            __builtin_prefetch(dict + (size_t)(dBase + 16) * C_DIM + tid * 16, 0, 1);
#pragma unroll
        for (int i = 0; i < 16; ++i)      // coalesced: 256 threads cover one row
            dictLds[i * RSTR + tid] = dict[(size_t)(dBase + i) * C_DIM + tid];
        __syncthreads();

        if (tid < 16) {                   // per-tile ||d||^2
            float s = 0.f;
            for (int c = 0; c < C_DIM; ++c) {
                float v = dictLds[tid * RSTR + c];
                s += v * v;
            }
            dnLds[tid] = s;
        }
        __syncthreads();

        // ---- 16x16 dot tile via 64 x v_wmma_f32_16x16x4_f32
        v8f acc = {};
#pragma unroll 8
        for (int kt = 0; kt < K_STEPS; ++kt) {
            const int c0 = kt * 4 + khalf * 2;
            v2f a = *(const v2f*)(aRow + c0);
            v2f b = *(const v2f*)(bRow + c0);
            acc = __builtin_amdgcn_wmma_f32_16x16x4_f32(
                false, a, false, b, (short)0, acc, false, false);
        }

        const float dn = dnLds[m];
        const int   dg = dBase + m;
#pragma unroll
        for (int r = 0; r < 8; ++r) {
            float s = dn - 2.0f * acc[r];
            if (s < best[r] || (s == best[r] && dg < bidx[r])) {
                best[r] = s; bidx[r] = dg;
            }
        }
    }

    // ---- cross-lane argmin within each 16-lane half (wave32)
#pragma unroll
    for (int r = 0; r < 8; ++r) {
        float s  = best[r];
        int   i0 = bidx[r];
#pragma unroll
        for (int off = 1; off < 16; off <<= 1) {
            float s2 = __shfl_xor(s, off, 32);
            int   i2 = __shfl_xor(i0, off, 32);
            if (s2 < s || (s2 == s && i2 < i0)) { s = s2; i0 = i2; }
        }
        best[r] = s; bidx[r] = i0;
    }

    if ((lane & 15) == 0) {               // lanes 0 (M=0..7) and 16 (M=8..15)
#pragma unroll
        for (int r = 0; r < 8; ++r) {
            int p = wave * 16 + khalf * 8 + r;
            idxLds[p] = bidx[r];
            out[2 * NCHW + (size_t)(wgPixBase + p)] = (float)bidx[r];
        }
    }
    __syncthreads();

    // ---- gather embedded + passthrough (identical fwd), coalesced NCHW stores
    float* emb   = out;
    float* embpt = out + NCHW;
    for (int j = tid; j < PIX_PER_WG * C_DIM; j += THREADS) {
        int c = j >> 7;
        int p = j & (PIX_PER_WG - 1);
        float v = dict[(size_t)idxLds[p] * C_DIM + c];
        size_t o = (size_t)img * C_DIM * HW + (size_t)c * HW + hw0 + p;
        emb[o]   = v;
        embpt[o] = v;
    }
}

extern "C" void kernel_launch(void* const* d_in, const int* in_sizes, int n_in,
                              void* d_out, int out_size, void* d_ws, size_t ws_size,
                              hipStream_t stream) {
    const float* x    = (const float*)d_in[0];
    const float* dict = (const float*)d_in[1];
    float* out = (float*)d_out;
    (void)in_sizes; (void)n_in; (void)out_size; (void)d_ws; (void)ws_size;
    dim3 grid(M_TOTAL / PIX_PER_WG);      // 256 workgroups
    dim3 block(THREADS);                  // 8 waves (wave32)
    vq_wmma_kernel<<<grid, block, SMEM_BYTES, stream>>>(x, dict, out);
}